// CoxLoss_79663053406538
// MI455X (gfx1250) — compile-verified
//
#include <hip/hip_runtime.h>
#include <hip/hip_bf16.h>
#include <stdint.h>

// Cox partial-likelihood loss, sort-free bucket-partition formulation.
// N = 4M, intermediates ~35MB -> fully L2-resident on MI455X (192MB L2).
// rss(i) = sum_{t[j] >= t[i]} exp(p[j]) decomposes into (suffix over later
// buckets) + (tie-exact in-bucket suffix); loss = (Σ e·log(rss) − Σ e·p)/Σe.

#define NB      131072u   // 2^17 monotone time buckets (~32 elems each)
#define NCH     128       // scan chunks: NB / 1024
#define SCAN_T  1024
#define KMAX    256       // bucket elems staged in LDS (2KB/block -> max occupancy)

// ---------------- K1: bucket histogram + n_events + sum(e*p) partials ----------
__global__ __launch_bounds__(256) void cox_k1_hist(
    const float* __restrict__ pred, const float* __restrict__ times,
    const int* __restrict__ events, unsigned int* __restrict__ counts,
    float* __restrict__ partialEP, unsigned int* __restrict__ nEvents, int n)
{
  __shared__ float red[256];
  const int tid = threadIdx.x;
  const int i = blockIdx.x * 256 + tid;
  float ep = 0.f; int e = 0;
  if (i < n) {
    __builtin_prefetch(&times[i + 16384], 0, 0);   // global_prefetch_b8
    float t = times[i];
    unsigned int b = (unsigned int)(t * (float)NB);
    if (b >= NB) b = NB - 1u;
    atomicAdd(&counts[b], 1u);
    e = (events[i] != 0);
    ep = e ? pred[i] : 0.f;
  }
  red[tid] = ep; __syncthreads();
  for (int s = 128; s; s >>= 1) { if (tid < s) red[tid] += red[tid + s]; __syncthreads(); }
  if (tid == 0) partialEP[blockIdx.x] = red[0];
  unsigned long long m = __ballot(e);               // wave32: low 32 bits used
  if ((tid & 31) == 0) atomicAdd(nEvents, (unsigned int)__popcll(m));
}

// ---------------- hierarchical exclusive scan of counts -> offs ----------------
__global__ __launch_bounds__(256) void cox_scanA(
    const unsigned int* __restrict__ counts, unsigned int* __restrict__ chunkTot)
{
  __shared__ unsigned int red[256];
  const int tid = threadIdx.x;
  unsigned int s = 0;
  const unsigned int* c = counts + (size_t)blockIdx.x * SCAN_T;
  for (int i = tid; i < SCAN_T; i += 256) s += c[i];
  red[tid] = s; __syncthreads();
  for (int st = 128; st; st >>= 1) { if (tid < st) red[tid] += red[tid + st]; __syncthreads(); }
  if (tid == 0) chunkTot[blockIdx.x] = red[0];
}

__global__ __launch_bounds__(NCH) void cox_scanB(
    const unsigned int* __restrict__ chunkTot, unsigned int* __restrict__ chunkBase,
    unsigned int* __restrict__ offs, int nb)
{
  __shared__ unsigned int t[NCH];
  const int tid = threadIdx.x;
  unsigned int v = chunkTot[tid];
  t[tid] = v; __syncthreads();
  for (int off = 1; off < NCH; off <<= 1) {
    unsigned int a = (tid >= off) ? t[tid - off] : 0u;
    __syncthreads(); t[tid] += a; __syncthreads();
  }
  chunkBase[tid] = t[tid] - v;                 // exclusive
  if (tid == NCH - 1) offs[nb] = t[NCH - 1];   // total == n
}

__global__ __launch_bounds__(SCAN_T) void cox_scanC(
    const unsigned int* __restrict__ counts, const unsigned int* __restrict__ chunkBase,
    unsigned int* __restrict__ offs)
{
  __shared__ unsigned int t[SCAN_T];
  const int tid = threadIdx.x;
  const size_t base = (size_t)blockIdx.x * SCAN_T;
  unsigned int v = counts[base + tid];
  t[tid] = v; __syncthreads();
  for (int off = 1; off < SCAN_T; off <<= 1) {
    unsigned int a = (tid >= off) ? t[tid - off] : 0u;
    __syncthreads(); t[tid] += a; __syncthreads();
  }
  offs[base + tid] = chunkBase[blockIdx.x] + (t[tid] - v);
}

// ---------------- K3: scatter 8B pairs (key|event, bits(exp(p))) --------------
__global__ __launch_bounds__(256) void cox_k3_scatter(
    const float* __restrict__ pred, const float* __restrict__ times,
    const int* __restrict__ events, const unsigned int* __restrict__ offs,
    unsigned int* __restrict__ cursors, uint2* __restrict__ spair, int n)
{
  const int i = blockIdx.x * 256 + threadIdx.x;
  if (i >= n) return;
  float t = times[i];
  unsigned int key = __float_as_uint(t);             // t>=0: bit order == value order
  unsigned int b = (unsigned int)(t * (float)NB);
  if (b >= NB) b = NB - 1u;
  unsigned int pos = offs[b] + atomicAdd(&cursors[b], 1u);
  uint2 v;
  v.x = key | (events[i] ? 0x80000000u : 0u);        // sign bit free (t in [0,1))
  v.y = __float_as_uint(expf(pred[i]));
  spair[pos] = v;                                    // single global_store_b64
}

// ------- sufA (fused per-bucket exp-sum + chunk totals, 4 buckets/thread) -----
__global__ __launch_bounds__(256) void cox_sufA(
    const unsigned int* __restrict__ offs, const uint2* __restrict__ spair,
    float* __restrict__ bucketSum, float* __restrict__ chunkTotF)
{
  __shared__ float red[256];
  const int tid = threadIdx.x;
  const int base = blockIdx.x * SCAN_T;
  float s = 0.f;
  for (int q = 0; q < 4; ++q) {                      // contiguous buckets per thread
    int b = base + tid * 4 + q;
    unsigned int s0 = offs[b], s1 = offs[b + 1];
    float bs = 0.f;
    for (unsigned int u = s0; u < s1; ++u) bs += __uint_as_float(spair[u].y);
    bucketSum[b] = bs;
    s += bs;
  }
  red[tid] = s; __syncthreads();
  for (int st = 128; st; st >>= 1) { if (tid < st) red[tid] += red[tid + st]; __syncthreads(); }
  if (tid == 0) chunkTotF[blockIdx.x] = red[0];
}

// suffixExcl[b] = Σ bucketSum[b'] over b' > b, built high->low end-to-end so the
// small suffixes (largest times, smallest risk sets) stay accurate.
__global__ __launch_bounds__(NCH) void cox_sufB(
    const float* __restrict__ chunkTotF, float* __restrict__ chunkCarryF)
{
  __shared__ float t[NCH];
  const int tid = threadIdx.x;
  float v = chunkTotF[tid];
  t[NCH - 1 - tid] = v; __syncthreads();       // reverse
  for (int off = 1; off < NCH; off <<= 1) {
    float a = (tid >= off) ? t[tid - off] : 0.f;
    __syncthreads(); t[tid] += a; __syncthreads();
  }
  chunkCarryF[tid] = t[NCH - 1 - tid] - v;     // Σ chunks strictly above
}

__global__ __launch_bounds__(SCAN_T) void cox_sufC(
    const float* __restrict__ bucketSum, const float* __restrict__ chunkCarryF,
    float* __restrict__ suffixExcl)
{
  __shared__ float t[SCAN_T];
  const int tid = threadIdx.x;
  const size_t base = (size_t)blockIdx.x * SCAN_T;
  float v = bucketSum[base + tid];
  t[SCAN_T - 1 - tid] = v; __syncthreads();    // reverse into LDS
  for (int off = 1; off < SCAN_T; off <<= 1) {
    float a = (tid >= off) ? t[tid - off] : 0.f;
    __syncthreads(); t[tid] += a; __syncthreads();
  }
  float rincl = t[SCAN_T - 1 - tid];           // Σ over [tid..end] of chunk
  suffixExcl[base + tid] = chunkCarryF[blockIdx.x] + (rincl - v);
}

// ---------------- K6: per-bucket tie-exact log(rss) for events ----------------
// One wave32 per bucket; 8B pairs staged into LDS with one CDNA5 async b64 copy
// per element (ASYNCcnt-tracked), consumed as single ds_load_b64 probes.
__global__ __launch_bounds__(32) void cox_k6_bucket(
    const unsigned int* __restrict__ offs, const uint2* __restrict__ spair,
    const float* __restrict__ suffixExcl, float* __restrict__ partialLOG)
{
  __shared__ uint2 lp[KMAX];                   // 2KB
  const int b = blockIdx.x;
  const int lane = threadIdx.x;
  const unsigned int start = offs[b];
  const int k = (int)(offs[b + 1] - start);
  float acc = 0.f;
  if (k > 0) {
    const uint2* pp;
    if (k <= KMAX) {
      for (int i = lane; i < k; i += 32) {
        unsigned int ldsA = (unsigned int)(uintptr_t)(&lp[i]);
        unsigned long long g = (unsigned long long)(uintptr_t)(spair + start + i);
        asm volatile("global_load_async_to_lds_b64 %0, %1, off"
                     :: "v"(ldsA), "v"(g) : "memory");
      }
      asm volatile("s_wait_asynccnt 0" ::: "memory");   // single wave: no barrier needed
      pp = lp;
    } else {                                   // astronomically rare overflow bucket
      pp = spair + start;
    }
    const float sfx = suffixExcl[b];           // Σ exp over all strictly-later buckets
    for (int i = lane; i < k; i += 32) {
      uint2 q = pp[i];
      if (q.x & 0x80000000u) {                 // event sample
        unsigned int mk = q.x & 0x7fffffffu;
        float s = 0.f;
        for (int j = 0; j < k; ++j) {
          uint2 r = pp[j];
          if ((r.x & 0x7fffffffu) >= mk) s += __uint_as_float(r.y);  // ties + self
        }
        acc += logf(sfx + s);
      }
    }
  }
  for (int off = 16; off; off >>= 1) acc += __shfl_down(acc, off);
  if (lane == 0) partialLOG[b] = acc;          // written for every bucket (incl. empty)
}

// ---------------- K7: final deterministic reduction ----------------
__global__ __launch_bounds__(1024) void cox_k7_final(
    const float* __restrict__ partialEP, int nep,
    const float* __restrict__ partialLOG, int nlog,
    const unsigned int* __restrict__ nEvents, float* __restrict__ out)
{
  __shared__ float red[1024];
  const int tid = threadIdx.x;
  float sep = 0.f, slg = 0.f;
  for (int i = tid; i < nep;  i += 1024) sep += partialEP[i];
  for (int i = tid; i < nlog; i += 1024) slg += partialLOG[i];
  red[tid] = slg - sep; __syncthreads();
  for (int s = 512; s; s >>= 1) { if (tid < s) red[tid] += red[tid + s]; __syncthreads(); }
  if (tid == 0) out[0] = red[0] / (float)(*nEvents);
}

extern "C" void kernel_launch(void* const* d_in, const int* in_sizes, int n_in,
                              void* d_out, int out_size, void* d_ws, size_t ws_size,
                              hipStream_t stream)
{
  (void)n_in; (void)out_size; (void)ws_size;
  const float* pred  = (const float*)d_in[0];
  const float* times = (const float*)d_in[1];
  const int*   evts  = (const int*)d_in[2];
  float* out = (float*)d_out;
  const int n = in_sizes[0];
  const int nb1 = (n + 255) / 256;

  // workspace carve-up (256B aligned), ~35MB total
  char* w = (char*)d_ws;
  auto carve = [&](size_t bytes) { void* p = (void*)w; w += (bytes + 255) & ~(size_t)255; return p; };
  unsigned int* counts      = (unsigned int*)carve((size_t)NB * 4);
  unsigned int* cursors     = (unsigned int*)carve((size_t)NB * 4);
  unsigned int* offs        = (unsigned int*)carve(((size_t)NB + 1) * 4);
  float*        bucketSum   = (float*)carve((size_t)NB * 4);
  float*        suffixExcl  = (float*)carve((size_t)NB * 4);
  float*        partialLOG  = (float*)carve((size_t)NB * 4);
  float*        partialEP   = (float*)carve((size_t)nb1 * 4);
  unsigned int* chunkTot    = (unsigned int*)carve((size_t)NCH * 4);
  unsigned int* chunkBase   = (unsigned int*)carve((size_t)NCH * 4);
  float*        chunkTotF   = (float*)carve((size_t)NCH * 4);
  float*        chunkCarryF = (float*)carve((size_t)NCH * 4);
  unsigned int* nEvents     = (unsigned int*)carve(4);
  uint2*        spair       = (uint2*)carve((size_t)n * 8);

  hipMemsetAsync(counts,  0, (size_t)NB * 4, stream);
  hipMemsetAsync(cursors, 0, (size_t)NB * 4, stream);
  hipMemsetAsync(nEvents, 0, 4, stream);

  cox_k1_hist   <<<nb1, 256, 0, stream>>>(pred, times, evts, counts, partialEP, nEvents, n);
  // exclusive scan counts -> offs (hierarchical, all phases parallel)
  cox_scanA     <<<NCH, 256, 0, stream>>>(counts, chunkTot);
  cox_scanB     <<<1, NCH, 0, stream>>>(chunkTot, chunkBase, offs, (int)NB);
  cox_scanC     <<<NCH, SCAN_T, 0, stream>>>(counts, chunkBase, offs);
  cox_k3_scatter<<<nb1, 256, 0, stream>>>(pred, times, evts, offs, cursors, spair, n);
  // per-bucket sums + reverse exclusive suffix scan (hierarchical)
  cox_sufA      <<<NCH, 256, 0, stream>>>(offs, spair, bucketSum, chunkTotF);
  cox_sufB      <<<1, NCH, 0, stream>>>(chunkTotF, chunkCarryF);
  cox_sufC      <<<NCH, SCAN_T, 0, stream>>>(bucketSum, chunkCarryF, suffixExcl);
  cox_k6_bucket <<<NB, 32, 0, stream>>>(offs, spair, suffixExcl, partialLOG);
  cox_k7_final  <<<1, 1024, 0, stream>>>(partialEP, nb1, partialLOG, (int)NB, nEvents, out);
}